// GraphNetWithSAGPooling_70626442215575
// MI455X (gfx1250) — compile-verified
//
#include <hip/hip_runtime.h>
#include <hip/hip_bf16.h>
#include <math.h>

// ---------------- problem constants ----------------
#define G_     64
#define NPG_   1600
#define N_     (G_ * NPG_)      // 102400
#define EPG_   51200
#define E_     (G_ * EPG_)      // 3276800
#define F_IN_  512
#define H_OUT_ 512
#define K_     1280             // kept nodes per graph
#define NK_    (G_ * K_)        // 81920
#define SORTN_ 2048
#define RPB_   16               // kept rows per gather block (K_ % RPB_ == 0)

typedef float v2f __attribute__((ext_vector_type(2)));
typedef float v8f __attribute__((ext_vector_type(8)));

#define CDIV(a, b) (((a) + (b) - 1) / (b))

// ---------------- utility init kernels ----------------
__global__ void zero_f32_kernel(float* p, int n) {
  int i = blockIdx.x * blockDim.x + threadIdx.x;
  if (i < n) p[i] = 0.0f;
}
__global__ void fill_i32_kernel(int* p, int v, int n) {
  int i = blockIdx.x * blockDim.x + threadIdx.x;
  if (i < n) p[i] = v;
}

// ---------------- conv1 degree / norm ----------------
__global__ void deg1_kernel(const int* __restrict__ dst, float* __restrict__ deg1) {
  int e = blockIdx.x * blockDim.x + threadIdx.x;
  if (e < E_) atomicAdd(&deg1[dst[e]], 1.0f);
}
__global__ void dis1_kernel(const float* __restrict__ deg1, float* __restrict__ dis1) {
  int n = blockIdx.x * blockDim.x + threadIdx.x;
  if (n < N_) dis1[n] = rsqrtf(deg1[n] + 1.0f);
}

// ---------------- GEMM1: h = x @ W1  (WMMA f32 16x16x4) ----------------
// x: [N,512] row-major, W1: [512,16] row-major, h: [N,16]
__global__ __launch_bounds__(256) void gemm_xw1_kernel(const float* __restrict__ x,
                                                       const float* __restrict__ W1,
                                                       float* __restrict__ h) {
  const int lane = threadIdx.x & 31;
  const int wave = threadIdx.x >> 5;
  const int tile = blockIdx.x * 8 + wave;   // 6400 M-tiles total
  const int row0 = tile * 16;
  const int half = lane >> 4;               // 0 or 1
  const int l15  = lane & 15;
  const int m    = row0 + l15;              // A row held by this lane
  v8f c = {};
  for (int k0 = 0; k0 < F_IN_; k0 += 4) {
    const int ka = k0 + half * 2;
    v2f a; a.x = x[m * F_IN_ + ka];         a.y = x[m * F_IN_ + ka + 1];
    v2f b; b.x = W1[(ka + 0) * 16 + l15];   b.y = W1[(ka + 1) * 16 + l15];
    c = __builtin_amdgcn_wmma_f32_16x16x4_f32(false, a, false, b, (short)0, c,
                                              false, false);
  }
  const int mb = row0 + half * 8;
#pragma unroll
  for (int v = 0; v < 8; ++v) h[(mb + v) * 16 + l15] = c[v];
}

// ---------------- conv1 edge aggregation (normalized) ----------------
__global__ void agg1_kernel(const int* __restrict__ src, const int* __restrict__ dst,
                            const float* __restrict__ h, const float* __restrict__ dis1,
                            float* __restrict__ agg1) {
  int gid = blockIdx.x * blockDim.x + threadIdx.x;   // E*16 threads
  if (gid < E_ * 16) {
    int e = gid >> 4, cidx = gid & 15;
    int s = src[e], d = dst[e];
    float norm = dis1[s] * dis1[d];
    atomicAdd(&agg1[d * 16 + cidx], h[s * 16 + cidx] * norm);
  }
}

__global__ void fin1_kernel(const float* __restrict__ agg1, const float* __restrict__ h,
                            const float* __restrict__ dis1, const float* __restrict__ b1,
                            float* __restrict__ h1) {
  int gid = blockIdx.x * blockDim.x + threadIdx.x;
  if (gid < N_ * 16) {
    int node = gid >> 4, cidx = gid & 15;
    float d = dis1[node];
    float v = agg1[gid] + h[gid] * d * d + b1[cidx];
    h1[gid] = v > 0.0f ? v : 0.0f;
  }
}

// ---------------- SAG score: unweighted neighbor sum ----------------
__global__ void agg2_kernel(const int* __restrict__ src, const int* __restrict__ dst,
                            const float* __restrict__ h1, float* __restrict__ agg2) {
  int gid = blockIdx.x * blockDim.x + threadIdx.x;
  if (gid < E_ * 16) {
    int e = gid >> 4, cidx = gid & 15;
    atomicAdd(&agg2[dst[e] * 16 + cidx], h1[src[e] * 16 + cidx]);
  }
}

__global__ void score_kernel(const float* __restrict__ h1, const float* __restrict__ agg2,
                             const float* __restrict__ Wrel, const float* __restrict__ brel,
                             const float* __restrict__ Wroot, float* __restrict__ score) {
  int n = blockIdx.x * blockDim.x + threadIdx.x;
  if (n < N_) {
    float acc = brel[0];
#pragma unroll
    for (int c = 0; c < 16; ++c)
      acc += agg2[n * 16 + c] * Wrel[c] + h1[n * 16 + c] * Wroot[c];
    score[n] = tanhf(acc);
  }
}

// ---------------- per-graph top-K via LDS bitonic sort ----------------
__global__ __launch_bounds__(1024) void topk_kernel(const float* __restrict__ score,
                                                    int* __restrict__ perm,
                                                    float* __restrict__ vals,
                                                    int* __restrict__ inv) {
  __shared__ float s[SORTN_];
  __shared__ int   id[SORTN_];
  const int g = blockIdx.x;
  const int t = threadIdx.x;
  for (int i = t; i < SORTN_; i += 1024) {
    if (i < NPG_) { s[i] = score[g * NPG_ + i]; }
    else          { s[i] = -__builtin_inff(); }
    id[i] = i;
  }
  __syncthreads();
  for (int k = 2; k <= SORTN_; k <<= 1) {
    for (int j = k >> 1; j > 0; j >>= 1) {
      for (int i = t; i < SORTN_; i += 1024) {
        int ixj = i ^ j;
        if (ixj > i) {
          bool desc = ((i & k) == 0);
          float a = s[i], b = s[ixj];
          int ia = id[i], ib = id[ixj];
          bool a_first = (a > b) || (a == b && ia < ib);   // deterministic tie-break
          bool doswap = desc ? !a_first : a_first;
          if (doswap) { s[i] = b; s[ixj] = a; id[i] = ib; id[ixj] = ia; }
        }
      }
      __syncthreads();
    }
  }
  for (int i = t; i < K_; i += 1024) {
    int node = g * NPG_ + id[i];
    int krow = g * K_ + i;
    perm[krow] = node;
    vals[krow] = s[i];
    inv[node]  = krow;
  }
}

// ---------------- conv2: filtered degree / CSR build ----------------
__global__ void cnt2_kernel(const int* __restrict__ src, const int* __restrict__ dst,
                            const int* __restrict__ inv, int* __restrict__ cnt2) {
  int e = blockIdx.x * blockDim.x + threadIdx.x;
  if (e < E_) {
    int is = inv[src[e]], idd = inv[dst[e]];
    if (is >= 0 && idd >= 0) atomicAdd(&cnt2[idd], 1);
  }
}
__global__ void dis2_kernel(const int* __restrict__ cnt2, float* __restrict__ dis2) {
  int i = blockIdx.x * blockDim.x + threadIdx.x;
  if (i < NK_) dis2[i] = rsqrtf((float)cnt2[i] + 1.0f);
}

// single-block exclusive scan over NK_ counts -> offsets (+ cursor copy)
__global__ __launch_bounds__(1024) void scan_kernel(const int* __restrict__ cnt,
                                                    int* __restrict__ off,
                                                    int* __restrict__ cur) {
  __shared__ int part[1024];
  const int t = threadIdx.x;
  const int CH = NK_ / 1024;      // 80
  const int base = t * CH;
  int s = 0;
  for (int i = 0; i < CH; ++i) s += cnt[base + i];
  part[t] = s;
  __syncthreads();
  int v = s;
  for (int d = 1; d < 1024; d <<= 1) {
    int add = (t >= d) ? part[t - d] : 0;
    __syncthreads();
    v += add;
    part[t] = v;
    __syncthreads();
  }
  int run = v - s;                // exclusive prefix of this chunk
  for (int i = 0; i < CH; ++i) {
    off[base + i] = run; cur[base + i] = run;
    run += cnt[base + i];
  }
  if (t == 1023) off[NK_] = run;
}

__global__ void fillcsr_kernel(const int* __restrict__ src, const int* __restrict__ dst,
                               const int* __restrict__ inv, int* __restrict__ cur,
                               int* __restrict__ csr) {
  int e = blockIdx.x * blockDim.x + threadIdx.x;
  if (e < E_) {
    int is = inv[src[e]], idd = inv[dst[e]];
    if (is >= 0 && idd >= 0) { int p = atomicAdd(&cur[idd], 1); csr[p] = is; }
  }
}

// ---------------- xk = h1[perm] * vals ----------------
__global__ void xk_kernel(const float* __restrict__ h1, const int* __restrict__ perm,
                          const float* __restrict__ vals, float* __restrict__ xk) {
  int gid = blockIdx.x * blockDim.x + threadIdx.x;
  if (gid < NK_ * 16) {
    int i = gid >> 4, c = gid & 15;
    xk[gid] = h1[perm[i] * 16 + c] * vals[i];
  }
}

// ---------------- GEMM2: hk = (xk @ W2) * dis2[row]  (WMMA f32 16x16x4) --------
// xk: [NK,16], W2: [16,512], hk: [NK,512] scaled by dis2[row]
__global__ __launch_bounds__(256) void gemm_xkw2_kernel(const float* __restrict__ xk,
                                                        const float* __restrict__ W2,
                                                        const float* __restrict__ dis2,
                                                        float* __restrict__ hk) {
  const int lane = threadIdx.x & 31;
  const int wave = threadIdx.x >> 5;
  const int tile = blockIdx.x * 8 + wave;   // 5120 mtiles * 32 ntiles
  const int mt = tile >> 5, nt = tile & 31;
  const int row0 = mt * 16, col0 = nt * 16;
  const int half = lane >> 4;
  const int l15  = lane & 15;
  const int m    = row0 + l15;
  v8f c = {};
#pragma unroll
  for (int k0 = 0; k0 < 16; k0 += 4) {
    const int ka = k0 + half * 2;
    v2f a; a.x = xk[m * 16 + ka];                 a.y = xk[m * 16 + ka + 1];
    v2f b; b.x = W2[(ka + 0) * H_OUT_ + col0 + l15];
           b.y = W2[(ka + 1) * H_OUT_ + col0 + l15];
    c = __builtin_amdgcn_wmma_f32_16x16x4_f32(false, a, false, b, (short)0, c,
                                              false, false);
  }
  const int n  = col0 + l15;
  const int mb = row0 + half * 8;
#pragma unroll
  for (int v = 0; v < 8; ++v)
    hk[(mb + v) * H_OUT_ + n] = c[v] * dis2[mb + v];
}

// ---------------- conv2 gather + ReLU + fused mean-pool partials ----------------
// hk holds rows pre-scaled by dis2[row].  h2[i] = relu(dis2[i]*(sum_e hk[src]+hk[i]) + b2)
__global__ __launch_bounds__(512) void gather_kernel(const float* __restrict__ hk,
                                                     const float* __restrict__ dis2,
                                                     const int* __restrict__ off,
                                                     const int* __restrict__ csr,
                                                     const float* __restrict__ b2,
                                                     float* __restrict__ out) {
  const int bpg   = K_ / RPB_;                 // blocks per graph (80)
  const int g     = blockIdx.x / bpg;
  const int chunk = blockIdx.x % bpg;
  const int row0  = g * K_ + chunk * RPB_;
  const int c     = threadIdx.x;               // 0..511 column
  const float bc  = b2[c];
  float gacc = 0.0f;
  for (int r = 0; r < RPB_; ++r) {
    const int i  = row0 + r;
    const int e0 = off[i], e1 = off[i + 1];
    float acc = hk[(size_t)i * H_OUT_ + c];    // self-loop term (pre-scaled)
    for (int e = e0; e < e1; ++e) {
      const int srow = csr[e];                 // scalar broadcast across block
      acc += hk[(size_t)srow * H_OUT_ + c];    // L2-resident coalesced gather
    }
    float h2 = dis2[i] * acc + bc;
    gacc += (h2 > 0.0f) ? h2 : 0.0f;
  }
  atomicAdd(&out[g * H_OUT_ + c], gacc * (1.0f / (float)K_));
}

// ---------------- host launcher ----------------
extern "C" void kernel_launch(void* const* d_in, const int* in_sizes, int n_in,
                              void* d_out, int out_size, void* d_ws, size_t ws_size,
                              hipStream_t stream) {
  const float* x     = (const float*)d_in[0];
  const int*   ei    = (const int*)d_in[1];
  const int*   src   = ei;
  const int*   dst   = ei + E_;
  const float* W1    = (const float*)d_in[3];
  const float* b1    = (const float*)d_in[4];
  const float* Wrel  = (const float*)d_in[5];
  const float* brel  = (const float*)d_in[6];
  const float* Wroot = (const float*)d_in[7];
  const float* W2    = (const float*)d_in[8];
  const float* b2    = (const float*)d_in[9];
  float*       out   = (float*)d_out;

  // carve workspace (256B aligned)
  char* p = (char*)d_ws;
  auto carve = [&](size_t bytes) -> void* {
    char* r = p;
    p += (bytes + 255) & ~(size_t)255;
    return (void*)r;
  };
  float* deg1 = (float*)carve((size_t)N_ * 4);
  float* dis1 = (float*)carve((size_t)N_ * 4);
  float* h    = (float*)carve((size_t)N_ * 16 * 4);
  float* agg1 = (float*)carve((size_t)N_ * 16 * 4);
  float* h1   = (float*)carve((size_t)N_ * 16 * 4);
  float* agg2 = (float*)carve((size_t)N_ * 16 * 4);
  float* scr  = (float*)carve((size_t)N_ * 4);
  int*   perm = (int*)carve((size_t)NK_ * 4);
  float* vals = (float*)carve((size_t)NK_ * 4);
  int*   inv  = (int*)carve((size_t)N_ * 4);
  int*   cnt2 = (int*)carve((size_t)NK_ * 4);
  float* dis2 = (float*)carve((size_t)NK_ * 4);
  int*   off  = (int*)carve((size_t)(NK_ + 1) * 4);
  int*   cur  = (int*)carve((size_t)NK_ * 4);
  int*   csr  = (int*)carve((size_t)E_ * 4);
  float* xk   = (float*)carve((size_t)NK_ * 16 * 4);
  float* hk   = (float*)carve((size_t)NK_ * H_OUT_ * 4);

  const int T = 256;

  // init (atomically-accumulated buffers must be zeroed every call)
  zero_f32_kernel<<<CDIV(N_, T), T, 0, stream>>>(deg1, N_);
  zero_f32_kernel<<<CDIV(N_ * 16, T), T, 0, stream>>>(agg1, N_ * 16);
  zero_f32_kernel<<<CDIV(N_ * 16, T), T, 0, stream>>>(agg2, N_ * 16);
  zero_f32_kernel<<<CDIV(G_ * H_OUT_, T), T, 0, stream>>>(out, G_ * H_OUT_);
  fill_i32_kernel<<<CDIV(N_, T), T, 0, stream>>>(inv, -1, N_);
  fill_i32_kernel<<<CDIV(NK_, T), T, 0, stream>>>(cnt2, 0, NK_);

  // conv1
  deg1_kernel<<<CDIV(E_, T), T, 0, stream>>>(dst, deg1);
  dis1_kernel<<<CDIV(N_, T), T, 0, stream>>>(deg1, dis1);
  gemm_xw1_kernel<<<(N_ / 16) / 8, T, 0, stream>>>(x, W1, h);
  agg1_kernel<<<CDIV(E_ * 16, T), T, 0, stream>>>(src, dst, h, dis1, agg1);
  fin1_kernel<<<CDIV(N_ * 16, T), T, 0, stream>>>(agg1, h, dis1, b1, h1);

  // SAG score + top-k
  agg2_kernel<<<CDIV(E_ * 16, T), T, 0, stream>>>(src, dst, h1, agg2);
  score_kernel<<<CDIV(N_, T), T, 0, stream>>>(h1, agg2, Wrel, brel, Wroot, scr);
  topk_kernel<<<G_, 1024, 0, stream>>>(scr, perm, vals, inv);

  // conv2 structure
  cnt2_kernel<<<CDIV(E_, T), T, 0, stream>>>(src, dst, inv, cnt2);
  dis2_kernel<<<CDIV(NK_, T), T, 0, stream>>>(cnt2, dis2);
  scan_kernel<<<1, 1024, 0, stream>>>(cnt2, off, cur);
  fillcsr_kernel<<<CDIV(E_, T), T, 0, stream>>>(src, dst, inv, cur, csr);

  // conv2 compute
  xk_kernel<<<CDIV(NK_ * 16, T), T, 0, stream>>>(h1, perm, vals, xk);
  gemm_xkw2_kernel<<<((NK_ / 16) * (H_OUT_ / 16)) / 8, T, 0, stream>>>(xk, W2, dis2, hk);
  gather_kernel<<<G_ * (K_ / RPB_), 512, 0, stream>>>(hk, dis2, off, csr, b2, out);
}